// SAGE_OneHot_hetero_42150809043600
// MI455X (gfx1250) — compile-verified
//
#include <hip/hip_runtime.h>

#define N_NODES 100000
#define IN_DIM  64
#define HIDDEN  128
#define N_EDGES 1600000

typedef float v2f __attribute__((ext_vector_type(2)));
typedef float v8f __attribute__((ext_vector_type(8)));

// ---------------------------------------------------------------- utilities
__global__ void zero_f32_kernel(float* __restrict__ p, int n) {
    int i = blockIdx.x * blockDim.x + threadIdx.x;
    if (i < n) p[i] = 0.0f;
}

// degree count: cnt[dst[e]] += 1
__global__ void degree_kernel(const long long* __restrict__ dst,
                              float* __restrict__ cnt) {
    int e = blockIdx.x * blockDim.x + threadIdx.x;
    if (e < N_EDGES) atomicAdd(&cnt[(int)dst[e]], 1.0f);
}

// scatter-add: agg[dst[e], :] += feat[src[e], :]
// One block of 256 threads handles EPB = 256/D edges; D consecutive lanes
// stream one feature row (coalesced gather of a 256B/512B row).
template <int D>
__global__ void scatter_kernel(const float* __restrict__ feat,
                               const long long* __restrict__ src,
                               const long long* __restrict__ dst,
                               float* __restrict__ agg) {
    constexpr int EPB = 256 / D;
    __shared__ int ssrc[EPB];
    __shared__ int sdst[EPB];
    const int t  = threadIdx.x;
    const int le = t / D;          // local edge within block
    const int f  = t % D;          // feature index
    const long long e = (long long)blockIdx.x * EPB + le;  // grid is exact
    if (f == 0) {
        ssrc[le] = (int)src[e];
        sdst[le] = (int)dst[e];
    }
    __syncthreads();
    const int s = ssrc[le];
    const int d = sdst[le];
    atomicAdd(&agg[(long long)d * D + f], feat[(long long)s * D + f]);
}

// in-place mean: agg[i,:] /= max(cnt[i], 1)
__global__ void mean_kernel(float* __restrict__ agg,
                            const float* __restrict__ cnt, int D) {
    long long i = (long long)blockIdx.x * blockDim.x + threadIdx.x;
    if (i < (long long)N_NODES * D) {
        int node = (int)(i / D);
        agg[i] /= fmaxf(cnt[node], 1.0f);
    }
}

// ------------------------------------------------------- fused SAGE GEMM
// H[M,HIDDEN] = act( A1[M,K]@W1[K,HIDDEN] + A2[M,K]@W2[K,HIDDEN] + b )
// One wave per 16x16 output tile, V_WMMA_F32_16X16X4_F32, K-step of 4.
// M = N_NODES = 6250*16 exactly; HIDDEN = 8*16 exactly -> no edge tiles.
template <int K, bool RELU>
__global__ __launch_bounds__(128)
void sage_gemm_wmma_kernel(const float* __restrict__ A1,
                           const float* __restrict__ W1,
                           const float* __restrict__ A2,
                           const float* __restrict__ W2,
                           const float* __restrict__ b,
                           float* __restrict__ H) {
    const int lane = threadIdx.x & 31;
    const int wave = threadIdx.x >> 5;
    const int tile = blockIdx.x * 4 + wave;        // 4 waves / block
    constexpr int TILES_N = HIDDEN / 16;           // 8
    const int tm = tile / TILES_N;
    const int tn = tile % TILES_N;
    const int m0 = tm * 16;
    const int n0 = tn * 16;

    // ISA 16x4 f32 A layout: lanes 0-15 -> M rows, VGPR pair holds K,K+1;
    // lanes 16-31 hold K+2,K+3 of the same M rows.
    const int mrow  = m0 + (lane & 15);
    const int khalf = (lane >> 4) * 2;             // 0 or 2
    const int ncol  = n0 + (lane & 15);

    v8f c = {};
    #pragma unroll 4
    for (int k = 0; k < K; k += 4) {
        const int kk = k + khalf;
        // A fragments: contiguous float2 in row-major A (8B aligned: K,kk even)
        const v2f a1 = *(const v2f*)(A1 + (long long)mrow * K + kk);
        const v2f a2 = *(const v2f*)(A2 + (long long)mrow * K + kk);
        // B fragments: rows kk and kk+1 of W at column ncol
        v2f w1, w2;
        w1.x = W1[(long long)kk * HIDDEN + ncol];
        w1.y = W1[(long long)(kk + 1) * HIDDEN + ncol];
        w2.x = W2[(long long)kk * HIDDEN + ncol];
        w2.y = W2[(long long)(kk + 1) * HIDDEN + ncol];
        c = __builtin_amdgcn_wmma_f32_16x16x4_f32(false, a1, false, w1,
                                                  (short)0, c, false, false);
        c = __builtin_amdgcn_wmma_f32_16x16x4_f32(false, a2, false, w2,
                                                  (short)0, c, false, false);
    }

    const float bv = b[ncol];
    // D layout: VGPR v -> M = m0 + v (+8 for upper half-wave), N = ncol
    const int rbase = m0 + ((lane >> 4) << 3);
    #pragma unroll
    for (int v = 0; v < 8; ++v) {
        float val = c[v] + bv;
        if (RELU) val = fmaxf(val, 0.0f);
        H[(long long)(rbase + v) * HIDDEN + ncol] = val;
    }
}

// out[i] = dot(h2[i,:], W_out) + b_out   (wave-per-node, shuffle reduce)
__global__ void out_kernel(const float* __restrict__ h2,
                           const float* __restrict__ Wout,
                           const float* __restrict__ bout,
                           float* __restrict__ out) {
    const int lane = threadIdx.x & 31;
    const int node = (int)((blockIdx.x * (long long)blockDim.x + threadIdx.x) >> 5);
    if (node >= N_NODES) return;
    float acc = 0.0f;
    #pragma unroll
    for (int f = lane; f < HIDDEN; f += 32)
        acc += h2[(long long)node * HIDDEN + f] * Wout[f];
    #pragma unroll
    for (int off = 16; off > 0; off >>= 1)
        acc += __shfl_down(acc, off, 32);
    if (lane == 0) out[node] = acc + bout[0];
}

// ------------------------------------------------------------------ driver
extern "C" void kernel_launch(void* const* d_in, const int* in_sizes, int n_in,
                              void* d_out, int out_size, void* d_ws, size_t ws_size,
                              hipStream_t stream) {
    (void)in_sizes; (void)n_in; (void)out_size; (void)ws_size;

    const float*     x     = (const float*)d_in[0];
    const long long* edges = (const long long*)d_in[1];   // int64 [2, E]
    const long long* src   = edges;
    const long long* dst   = edges + N_EDGES;
    const float* W_l1  = (const float*)d_in[2];
    const float* W_r1  = (const float*)d_in[3];
    const float* b1    = (const float*)d_in[4];
    const float* W_l2  = (const float*)d_in[5];
    const float* W_r2  = (const float*)d_in[6];
    const float* b2    = (const float*)d_in[7];
    const float* W_out = (const float*)d_in[8];
    const float* b_out = (const float*)d_in[9];

    // d_out layout: out [N], h1 [N*H], h2 [N*H]
    float* out = (float*)d_out;
    float* h1  = out + N_NODES;
    float* h2  = h1 + (long long)N_NODES * HIDDEN;

    // workspace: cnt [N] at 0, agg [N*HIDDEN] at 128-float-padded offset
    float* cnt = (float*)d_ws;
    float* agg = cnt + 100096;   // byte offset 400384, 256B aligned

    const int ZT = 256;
    const int gemm_blocks = (N_NODES / 16) * (HIDDEN / 16) / 4;  // 12500

    // ---- layer 1 ----
    zero_f32_kernel<<<(N_NODES + ZT - 1) / ZT, ZT, 0, stream>>>(cnt, N_NODES);
    zero_f32_kernel<<<(N_NODES * IN_DIM + ZT - 1) / ZT, ZT, 0, stream>>>(agg, N_NODES * IN_DIM);
    degree_kernel<<<N_EDGES / 256, 256, 0, stream>>>(dst, cnt);
    scatter_kernel<IN_DIM><<<N_EDGES / (256 / IN_DIM), 256, 0, stream>>>(x, src, dst, agg);
    mean_kernel<<<(N_NODES * IN_DIM + ZT - 1) / ZT, ZT, 0, stream>>>(agg, cnt, IN_DIM);
    sage_gemm_wmma_kernel<IN_DIM, true><<<gemm_blocks, 128, 0, stream>>>(
        agg, W_l1, x, W_r1, b1, h1);

    // ---- layer 2 ----
    zero_f32_kernel<<<(N_NODES * HIDDEN + ZT - 1) / ZT, ZT, 0, stream>>>(agg, N_NODES * HIDDEN);
    scatter_kernel<HIDDEN><<<N_EDGES / (256 / HIDDEN), 256, 0, stream>>>(h1, src, dst, agg);
    mean_kernel<<<(N_NODES * HIDDEN + ZT - 1) / ZT, ZT, 0, stream>>>(agg, cnt, HIDDEN);
    sage_gemm_wmma_kernel<HIDDEN, false><<<gemm_blocks, 128, 0, stream>>>(
        agg, W_l2, h1, W_r2, b2, h2);

    // ---- output head ----
    out_kernel<<<(N_NODES * 32 + 255) / 256, 256, 0, stream>>>(h2, W_out, b_out, out);
}